// GCN_4071628996848
// MI455X (gfx1250) — compile-verified
//
#include <hip/hip_runtime.h>
#include <math.h>

typedef __attribute__((ext_vector_type(2))) float v2f;
typedef __attribute__((ext_vector_type(8))) float v8f;

#define LN_EPS 1e-5f

// ---------------------------------------------------------------------------
// Degree computation
// ---------------------------------------------------------------------------
__global__ void degree_kernel(const int* __restrict__ src, const int* __restrict__ dst,
                              float* __restrict__ dout, float* __restrict__ din, int e) {
    int i = blockIdx.x * blockDim.x + threadIdx.x;
    if (i >= e) return;
    atomicAdd(dout + src[i], 1.0f);
    atomicAdd(din  + dst[i], 1.0f);
}

__global__ void invsqrt_deg_kernel(float* __restrict__ dout, float* __restrict__ din, int n) {
    int i = blockIdx.x * blockDim.x + threadIdx.x;
    if (i >= n) return;
    dout[i] = rsqrtf(fmaxf(dout[i], 1.0f));
    din[i]  = rsqrtf(fmaxf(din[i],  1.0f));
}

// ---------------------------------------------------------------------------
// LayerNorm statistics for the 768-wide input features (one wave32 per row)
// ---------------------------------------------------------------------------
__global__ void ln_stats_768_kernel(const float* __restrict__ feat,
                                    float* __restrict__ mu, float* __restrict__ rs, int n) {
    int row  = (blockIdx.x * blockDim.x + threadIdx.x) >> 5;
    int lane = threadIdx.x & 31;
    if (row >= n) return;
    const float* p = feat + (size_t)row * 768;
    float s = 0.f, s2 = 0.f;
    #pragma unroll
    for (int c = 0; c < 768; c += 32) {
        float v = p[c + lane];
        s += v; s2 += v * v;
    }
    #pragma unroll
    for (int off = 16; off > 0; off >>= 1) {
        s  += __shfl_xor(s,  off, 32);
        s2 += __shfl_xor(s2, off, 32);
    }
    if (lane == 0) {
        float m   = s  * (1.0f / 768.0f);
        float var = s2 * (1.0f / 768.0f) - m * m;
        mu[row] = m;
        rs[row] = rsqrtf(var + LN_EPS);
    }
}

// ---------------------------------------------------------------------------
// GEMM0, fused: C[N x 128] = (LN(feat)*g+b) * do_is  @  W0[768 x 128]
// One wave per 16-row strip; 8 independent f32 WMMA accumulators (16x128).
// A 16x4 f32 fragment: lanes 0-15 hold K=k,k+1 ; lanes 16-31 hold K=k+2,k+3.
// B 4x16 fragment mirrored: vgpr0 = rows {k, k+2}, vgpr1 = rows {k+1, k+3}.
// ---------------------------------------------------------------------------
__global__ void gemm0_ln_wmma_kernel(const float* __restrict__ feat,
                                     const float* __restrict__ mu, const float* __restrict__ rs,
                                     const float* __restrict__ do_is,
                                     const float* __restrict__ g, const float* __restrict__ b,
                                     const float* __restrict__ W,  // 768 x 128
                                     float* __restrict__ C, int M) {
    const int K = 768, F = 128;
    const int lane = threadIdx.x & 31;
    const int mt   = (blockIdx.x * blockDim.x + threadIdx.x) >> 5;
    if (mt * 16 >= M) return;
    const int half = lane >> 4;      // 0: K pair {0,1}, 1: K pair {2,3}
    const int l16  = lane & 15;
    const int arow = mt * 16 + l16;
    const int koff = half * 2;
    const float m_r  = mu[arow];
    const float rs_r = rs[arow];
    const float do_r = do_is[arow];
    const float* ap = feat + (size_t)arow * K + koff;   // 8B aligned (koff even)

    v8f acc[8];
    #pragma unroll
    for (int i = 0; i < 8; ++i) { v8f z = {}; acc[i] = z; }

    for (int k = 0; k < K; k += 4) {
        const int c0 = k + koff;
        const v2f x  = *(const v2f*)(ap + k);           // global_load_b64
        const v2f gg = *(const v2f*)(g + c0);
        const v2f bbv = *(const v2f*)(b + c0);
        v2f a;
        a.x = ((x.x - m_r) * rs_r * gg.x + bbv.x) * do_r;
        a.y = ((x.y - m_r) * rs_r * gg.y + bbv.y) * do_r;
        const float* wrow = W + (size_t)c0 * F;  // row c0; row c0+1 = +F
        #pragma unroll
        for (int ft = 0; ft < 8; ++ft) {
            const int bcol = ft * 16 + l16;
            v2f bb;
            bb.x = wrow[bcol];
            bb.y = wrow[bcol + F];
            acc[ft] = __builtin_amdgcn_wmma_f32_16x16x4_f32(
                false, a, false, bb, (short)0, acc[ft], false, false);
        }
    }
    // C/D layout: VGPR r -> row m0+r (lanes 0-15) / m0+8+r (lanes 16-31), col = n0+l16
    const int crow = mt * 16 + half * 8;
    #pragma unroll
    for (int ft = 0; ft < 8; ++ft) {
        const int bcol = ft * 16 + l16;
        #pragma unroll
        for (int r = 0; r < 8; ++r)
            C[(size_t)(crow + r) * F + bcol] = acc[ft][r];
    }
}

// ---------------------------------------------------------------------------
// Generic strip GEMM: C[M x F] = A[M x K] @ W[K x F], FT = ceil(F/16) tiles.
// ---------------------------------------------------------------------------
template <int FT>
__global__ void gemm_wmma_strip_kernel(const float* __restrict__ A, const float* __restrict__ W,
                                       float* __restrict__ C, int M, int K, int F) {
    const int lane = threadIdx.x & 31;
    const int mt   = (blockIdx.x * blockDim.x + threadIdx.x) >> 5;
    if (mt * 16 >= M) return;
    const int half = lane >> 4;
    const int l16  = lane & 15;
    const int arow = mt * 16 + l16;
    const int koff = half * 2;
    const float* ap = A + (size_t)arow * K + koff;

    v8f acc[FT];
    #pragma unroll
    for (int i = 0; i < FT; ++i) { v8f z = {}; acc[i] = z; }

    for (int k = 0; k < K; k += 4) {
        const v2f a = *(const v2f*)(ap + k);            // global_load_b64
        const float* wrow = W + (size_t)(k + koff) * F;
        #pragma unroll
        for (int ft = 0; ft < FT; ++ft) {
            const int  bcol = ft * 16 + l16;
            const bool bv   = (bcol < F);
            const int  bc   = bv ? bcol : 0;
            v2f bb;
            bb.x = bv ? wrow[bc]     : 0.0f;
            bb.y = bv ? wrow[bc + F] : 0.0f;
            acc[ft] = __builtin_amdgcn_wmma_f32_16x16x4_f32(
                false, a, false, bb, (short)0, acc[ft], false, false);
        }
    }
    const int crow = mt * 16 + half * 8;
    #pragma unroll
    for (int ft = 0; ft < FT; ++ft) {
        const int bcol = ft * 16 + l16;
        if (bcol < F) {
            #pragma unroll
            for (int r = 0; r < 8; ++r)
                C[(size_t)(crow + r) * F + bcol] = acc[ft][r];
        }
    }
}

// ---------------------------------------------------------------------------
// Edge scatter-add, vectorized: out[dst[e], :] += x[src[e], :]
// One thread per (edge, float4-chunk); F = F4*4 features per row.
// Template constant F4 -> div/mod strength-reduced; gather is b128-coalesced.
// ---------------------------------------------------------------------------
template <int F4>
__global__ void scatter_add_vec4_kernel(const float* __restrict__ x,
                                        const int* __restrict__ src, const int* __restrict__ dst,
                                        float* __restrict__ out, int e) {
    long long t = (long long)blockIdx.x * blockDim.x + threadIdx.x;
    if (t >= (long long)e * F4) return;
    const int ed = (int)(t / F4);
    const int c4 = (int)(t % F4);
    const float4 v = ((const float4*)(x + (size_t)src[ed] * (F4 * 4)))[c4];
    float* o = out + (size_t)dst[ed] * (F4 * 4) + (size_t)c4 * 4;
    atomicAdd(o + 0, v.x);
    atomicAdd(o + 1, v.y);
    atomicAdd(o + 2, v.z);
    atomicAdd(o + 3, v.w);
}

// ---------------------------------------------------------------------------
// Fused:  h = relu(LN(agg * di_is)) * do_is     (128 wide, wave per row)
// ---------------------------------------------------------------------------
__global__ void post_ln_relu_128_kernel(const float* __restrict__ agg,
                                        const float* __restrict__ di_is,
                                        const float* __restrict__ do_is,
                                        const float* __restrict__ g, const float* __restrict__ b,
                                        float* __restrict__ out, int n) {
    int row  = (blockIdx.x * blockDim.x + threadIdx.x) >> 5;
    int lane = threadIdx.x & 31;
    if (row >= n) return;
    const float* p = agg + (size_t)row * 128;
    const float di = di_is[row];
    float v[4];
    float s = 0.f, s2 = 0.f;
    #pragma unroll
    for (int i = 0; i < 4; ++i) {
        v[i] = p[lane + 32 * i] * di;
        s += v[i]; s2 += v[i] * v[i];
    }
    #pragma unroll
    for (int off = 16; off > 0; off >>= 1) {
        s  += __shfl_xor(s,  off, 32);
        s2 += __shfl_xor(s2, off, 32);
    }
    const float m   = s  * (1.0f / 128.0f);
    const float rsv = rsqrtf(s2 * (1.0f / 128.0f) - m * m + LN_EPS);
    const float ds  = do_is[row];
    float* q = out + (size_t)row * 128;
    #pragma unroll
    for (int i = 0; i < 4; ++i) {
        const int c = lane + 32 * i;
        float y = (v[i] - m) * rsv * g[c] + b[c];
        q[c] = fmaxf(y, 0.0f) * ds;
    }
}

// ---------------------------------------------------------------------------
// Final: out = out * di_is[row] + b2[col]   (F = 40, compile-time)
// ---------------------------------------------------------------------------
__global__ void final_scale_bias_kernel(float* __restrict__ out,
                                        const float* __restrict__ di_is,
                                        const float* __restrict__ b2, int n) {
    const int F = 40;
    int t = blockIdx.x * blockDim.x + threadIdx.x;
    if (t >= n * F) return;
    out[t] = out[t] * di_is[t / F] + b2[t % F];
}

// ---------------------------------------------------------------------------
static inline int cdiv(long long a, long long b) { return (int)((a + b - 1) / b); }

extern "C" void kernel_launch(void* const* d_in, const int* in_sizes, int n_in,
                              void* d_out, int out_size, void* d_ws, size_t ws_size,
                              hipStream_t stream) {
    const float* feat = (const float*)d_in[0];
    const int*   src  = (const int*)d_in[1];
    const int*   dst  = (const int*)d_in[2];
    const float* W0   = (const float*)d_in[3];
    const float* W1   = (const float*)d_in[4];
    const float* W2   = (const float*)d_in[5];
    const float* b2   = (const float*)d_in[6];
    const float* g_in = (const float*)d_in[7];
    const float* b_in = (const float*)d_in[8];
    const float* g0   = (const float*)d_in[9];
    const float* b0   = (const float*)d_in[10];
    const float* g1   = (const float*)d_in[11];
    const float* b1   = (const float*)d_in[12];
    float* out = (float*)d_out;

    const int IN_F = 768, HID = 128, CLS = 40;
    const int n = in_sizes[0] / IN_F;   // 50000
    const int e = in_sizes[1];          // 800000

    // Workspace layout (floats): do_is | di_is | mu | rs | bufB | bufC | bufD | bufE
    float* ws    = (float*)d_ws;
    float* do_is = ws;
    float* di_is = ws + (size_t)n;
    float* mu    = ws + 2 * (size_t)n;
    float* rs    = ws + 3 * (size_t)n;
    float* bufB  = ws + 4 * (size_t)n;                 // N x 128
    float* bufC  = bufB + (size_t)n * HID;             // N x 128
    float* bufD  = bufC + (size_t)n * HID;             // N x 128
    float* bufE  = bufD + (size_t)n * HID;             // N x 40

    const int T = 256;

    // --- degrees ---
    hipMemsetAsync(do_is, 0, 2 * (size_t)n * sizeof(float), stream);
    degree_kernel<<<cdiv(e, T), T, 0, stream>>>(src, dst, do_is, di_is, e);
    invsqrt_deg_kernel<<<cdiv(n, T), T, 0, stream>>>(do_is, di_is, n);

    // --- input LayerNorm stats ---
    ln_stats_768_kernel<<<cdiv((long long)n * 32, T), T, 0, stream>>>(feat, mu, rs, n);

    // --- layer 0: (LN(feat)*do_is) @ W0  -> bufB ---
    const int mtiles = cdiv(n, 16);
    gemm0_ln_wmma_kernel<<<cdiv((long long)mtiles * 32, T), T, 0, stream>>>(
        feat, mu, rs, do_is, g_in, b_in, W0, bufB, n);

    // --- scatter over edges -> bufC (A0) ---
    hipMemsetAsync(bufC, 0, (size_t)n * HID * sizeof(float), stream);
    scatter_add_vec4_kernel<32><<<cdiv((long long)e * 32, T), T, 0, stream>>>(bufB, src, dst, bufC, e);

    // --- h1 = relu(LN(A0*di_is))*do_is -> bufD ---
    post_ln_relu_128_kernel<<<cdiv((long long)n * 32, T), T, 0, stream>>>(
        bufC, di_is, do_is, g0, b0, bufD, n);

    // --- layer 1: scatter first (W1 square), then GEMM ---
    hipMemsetAsync(bufB, 0, (size_t)n * HID * sizeof(float), stream);
    scatter_add_vec4_kernel<32><<<cdiv((long long)e * 32, T), T, 0, stream>>>(bufD, src, dst, bufB, e);
    gemm_wmma_strip_kernel<8><<<cdiv((long long)mtiles * 32, T), T, 0, stream>>>(
        bufB, W1, bufC, n, HID, HID);

    // --- h2 = relu(LN(Y1*di_is))*do_is -> bufD ---
    post_ln_relu_128_kernel<<<cdiv((long long)n * 32, T), T, 0, stream>>>(
        bufC, di_is, do_is, g1, b1, bufD, n);

    // --- layer 2: h2 @ W2 -> bufE (N x 40), scatter to out, scale + bias ---
    gemm_wmma_strip_kernel<3><<<cdiv((long long)mtiles * 32, T), T, 0, stream>>>(
        bufD, W2, bufE, n, HID, CLS);
    hipMemsetAsync(out, 0, (size_t)n * CLS * sizeof(float), stream);
    scatter_add_vec4_kernel<10><<<cdiv((long long)e * 10, T), T, 0, stream>>>(bufE, src, dst, out, e);
    final_scale_bias_kernel<<<cdiv((long long)n * CLS, T), T, 0, stream>>>(out, di_is, b2, n);
}